// GCNConvNetForRegression_89223650607760
// MI455X (gfx1250) — compile-verified
//
#include <hip/hip_runtime.h>
#include <hip/hip_bf16.h>
#include <stddef.h>

typedef __attribute__((ext_vector_type(2))) float v2f;
typedef __attribute__((ext_vector_type(8))) float v8f;

// ---------------------------------------------------------------------------
// Native device-scope f32 atomic add, pinned via inline asm so we are certain
// the scatter path uses GLOBAL_ATOMIC_ADD_F32 (no CAS-loop fallback).
// Non-returning -> tracked by STOREcnt; S_ENDPGM implicitly waits idle.
// ---------------------------------------------------------------------------
__device__ __forceinline__ void atomAddF(float* p, float v) {
  asm volatile("global_atomic_add_f32 %0, %1, off scope:SCOPE_DEV"
               :
               : "v"(p), "v"(v)
               : "memory");
}

// ---------------------------------------------------------------------------
// init: deg = 1.0 (self-loop contribution), agg = 0
// ---------------------------------------------------------------------------
__global__ void k_init(float* __restrict__ deg, float* __restrict__ agg,
                       int n, int tot) {
  int t = blockIdx.x * blockDim.x + threadIdx.x;
  if (t < tot) agg[t] = 0.0f;
  if (t < n)   deg[t] = 1.0f;
}

__global__ void k_zero(float* __restrict__ p, int tot) {
  int t = blockIdx.x * blockDim.x + threadIdx.x;
  if (t < tot) p[t] = 0.0f;
}

// ---------------------------------------------------------------------------
// degree accumulation over edge destinations
// ---------------------------------------------------------------------------
__global__ void k_degree(const int* __restrict__ dst, float* __restrict__ deg,
                         int E) {
  int t = blockIdx.x * blockDim.x + threadIdx.x;
  if (t < E) atomAddF(&deg[dst[t]], 1.0f);
}

__global__ void k_rsqrt(float* __restrict__ d, int n) {
  int t = blockIdx.x * blockDim.x + threadIdx.x;
  if (t < n) d[t] = rsqrtf(d[t]);
}

// ---------------------------------------------------------------------------
// Y[nrows x 16] = X[nrows x K] @ W[K x 16] via V_WMMA_F32_16X16X4_F32.
// One wave computes one 16x16 C tile; K must be a multiple of 4, nrows of 16.
//
// Layouts (CDNA5 ISA 7.12.2):
//   A 16x4 f32 : lane = kh*16 + m, VGPR j holds K = 2*kh + j  -> float2 load
//   B  4x16 f32: lane = kh*16 + n, VGPR j holds row K = 2*kh + j, col n
//   C 16x16 f32: lane = kh*16 + n, VGPR v holds row M = v + 8*kh, col n
// ---------------------------------------------------------------------------
__global__ void k_gemm16(const float* __restrict__ X, const float* __restrict__ W,
                         float* __restrict__ Y, int nrows, int K) {
  const int wave = threadIdx.x >> 5;
  const int lane = threadIdx.x & 31;
  const int tile = blockIdx.x * (blockDim.x >> 5) + wave;
  const int row0 = tile * 16;
  if (row0 >= nrows) return;                 // wave-uniform exit (EXEC all-1s)

  const int m  = lane & 15;                  // row (A) / col (B,C) within tile
  const int kh = lane >> 4;                  // K-half selector

  const float* ap = X + (size_t)(row0 + m) * K + 2 * kh;
  v8f acc = {};
  for (int kb = 0; kb < K; kb += 4) {
    v2f a = *(const v2f*)(ap + kb);
    const int kr = kb + 2 * kh;
    v2f b;
    b.x = W[(size_t)kr * 16 + m];
    b.y = W[(size_t)(kr + 1) * 16 + m];
    acc = __builtin_amdgcn_wmma_f32_16x16x4_f32(
        false, a, false, b, (short)0, acc, false, false);
  }
#pragma unroll
  for (int v = 0; v < 8; ++v) {
    Y[(size_t)(row0 + v + 8 * kh) * 16 + m] = acc[v];
  }
}

// ---------------------------------------------------------------------------
// edge scatter: agg[dst] += xw[src] * dinv[src]*dinv[dst]
// 4 threads per edge, float4 gather + 4 native f32 atomics each
// ---------------------------------------------------------------------------
__global__ void k_scatter(const int* __restrict__ src, const int* __restrict__ dst,
                          const float* __restrict__ dinv,
                          const float* __restrict__ xw,
                          float* __restrict__ agg, int E) {
  int t = blockIdx.x * blockDim.x + threadIdx.x;
  int e = t >> 2;
  if (e >= E) return;
  int q = (t & 3) << 2;
  int s = src[e];
  int d = dst[e];
  float nrm = dinv[s] * dinv[d];
  float4 v = *(const float4*)(xw + (size_t)s * 16 + q);
  float* out = agg + (size_t)d * 16 + q;
  atomAddF(out + 0, v.x * nrm);
  atomAddF(out + 1, v.y * nrm);
  atomAddF(out + 2, v.z * nrm);
  atomAddF(out + 3, v.w * nrm);
}

// ---------------------------------------------------------------------------
// finalize layer: agg = relu(agg + xw*dinv^2 + b)   (self-loop folded in)
// ---------------------------------------------------------------------------
__global__ void k_final1(float* __restrict__ agg, const float* __restrict__ xw,
                         const float* __restrict__ dinv,
                         const float* __restrict__ b, int tot) {
  int t = blockIdx.x * blockDim.x + threadIdx.x;
  if (t >= tot) return;
  int i = t >> 4, f = t & 15;
  float s = dinv[i];
  float v = agg[t] + xw[t] * s * s + b[f];
  agg[t] = fmaxf(v, 0.0f);
}

// ---------------------------------------------------------------------------
// finalize layer 2 + readout:
//   h = relu(agg + xw*dinv^2 + b2);  out = h . W_lin + b_lin
// ---------------------------------------------------------------------------
__global__ void k_final2(const float* __restrict__ agg, const float* __restrict__ xw,
                         const float* __restrict__ dinv,
                         const float* __restrict__ b2,
                         const float* __restrict__ Wl,
                         const float* __restrict__ bl,
                         float* __restrict__ out, int n) {
  int i = blockIdx.x * blockDim.x + threadIdx.x;
  if (i >= n) return;
  float s = dinv[i];
  float ss = s * s;
  float acc = 0.0f;
#pragma unroll
  for (int f = 0; f < 16; ++f) {
    float h = agg[(size_t)i * 16 + f] + xw[(size_t)i * 16 + f] * ss + b2[f];
    h = fmaxf(h, 0.0f);
    acc += h * Wl[f];
  }
  out[i] = acc + bl[0];
}

// ---------------------------------------------------------------------------
extern "C" void kernel_launch(void* const* d_in, const int* in_sizes, int n_in,
                              void* d_out, int out_size, void* d_ws, size_t ws_size,
                              hipStream_t stream) {
  const float* x   = (const float*)d_in[0];
  const int*   ei  = (const int*)d_in[1];   // [2, E] : row0 = src, row1 = dst
  const float* W1  = (const float*)d_in[2];
  const float* b1  = (const float*)d_in[3];
  const float* W2  = (const float*)d_in[4];
  const float* b2  = (const float*)d_in[5];
  const float* Wl  = (const float*)d_in[6];
  const float* bl  = (const float*)d_in[7];
  float* out = (float*)d_out;

  const int din = in_sizes[2] / 16;         // 128
  const int n   = in_sizes[0] / din;        // 100000
  const int E   = in_sizes[1] / 2;          // 3.2M
  const int tot = n * 16;

  float* ws   = (float*)d_ws;
  float* dinv = ws;                                       // n floats
  float* bufA = ws + (((size_t)n + 255) & ~(size_t)255);  // n*16 floats
  float* bufB = bufA + (size_t)n * 16;                    // n*16 floats

  const int T = 256;
  const int ntiles = (n + 15) / 16;
  const int gemmBlocks = (ntiles + 3) / 4;                // 4 waves / block

  // degree + norm
  k_init  <<<(tot + T - 1) / T, T, 0, stream>>>(dinv, bufB, n, tot);
  k_degree<<<(E + T - 1) / T,   T, 0, stream>>>(ei + E, dinv, E);
  k_rsqrt <<<(n + T - 1) / T,   T, 0, stream>>>(dinv, n);

  // layer 1: xw1 = x @ W1 ; scatter ; relu+bias (self-loop folded)
  k_gemm16 <<<gemmBlocks, 128, 0, stream>>>(x, W1, bufA, n, din);
  k_scatter<<<(4 * E + T - 1) / T, T, 0, stream>>>(ei, ei + E, dinv, bufA, bufB, E);
  k_final1 <<<(tot + T - 1) / T, T, 0, stream>>>(bufB, bufA, dinv, b1, tot);

  // layer 2: xw2 = h1 @ W2 ; scatter ; relu+bias fused with readout dot
  k_gemm16 <<<gemmBlocks, 128, 0, stream>>>(bufB, W2, bufA, n, 16);
  k_zero   <<<(tot + T - 1) / T, T, 0, stream>>>(bufB, tot);
  k_scatter<<<(4 * E + T - 1) / T, T, 0, stream>>>(ei, ei + E, dinv, bufA, bufB, E);
  k_final2 <<<(n + T - 1) / T, T, 0, stream>>>(bufB, bufA, dinv, b2, Wl, bl, out, n);
}